// Net_25254407701209
// MI455X (gfx1250) — compile-verified
//
#include <hip/hip_runtime.h>

typedef __attribute__((ext_vector_type(2))) float v2f;
typedef __attribute__((ext_vector_type(8))) float v8f;

#define LDS_STRIDE 260   // 260 % 64 == 4 -> conflict-free A-fragment reads

static __device__ __forceinline__ v8f wmma_f32_k4(v2f a, v2f b, v8f c) {
  // V_WMMA_F32_16X16X4_F32 : D = A(16x4) * B(4x16) + C, full f32
  return __builtin_amdgcn_wmma_f32_16x16x4_f32(false, a, false, b, (short)0, c,
                                               false, false);
}

__global__ __launch_bounds__(256, 1) void sindy_fused(
    const float* __restrict__ x,    const float* __restrict__ xdot,
    const float* __restrict__ w1,   const float* __restrict__ b1,
    const float* __restrict__ w2a,  const float* __restrict__ b2a,
    const float* __restrict__ w2b,  const float* __restrict__ b2b,
    const float* __restrict__ w3a,  const float* __restrict__ b3a,
    const float* __restrict__ w3s,  const float* __restrict__ b3s,
    const float* __restrict__ wout, const float* __restrict__ scalar,
    float* __restrict__ out_b, float* __restrict__ out_bdot,
    float* __restrict__ out_y, float* __restrict__ out_yy)
{
  __shared__ float lds_y[16 * LDS_STRIDE];  // y -> y2 -> y3
  __shared__ float lds_j[16 * LDS_STRIDE];  // ydot1 -> ydot2 -> ydot3
  __shared__ float red_b[256];
  __shared__ float red_d[256];

  const int tid  = threadIdx.x;
  const int wave = tid >> 5;
  const int lane = tid & 31;
  const int n    = lane & 15;   // N within 16-tile (also M index for A frags)
  const int h    = lane >> 4;   // half-wave -> K sub-pair selector
  const int row0 = blockIdx.x * 16;
  const int colbase = wave * 32;
  const int col0 = colbase + n;
  const int col1 = colbase + 16 + n;

  // ---------------- Phase 1: K=64 GEMMs (layer 1 + layer-3 skip path) -------
  v8f zz[2], ss[2], q3[2], qd[2];   // z, s=xdot@w1T, z2l3=x@w3sT, sdot=xdot@w3sT
  #pragma unroll
  for (int t = 0; t < 2; ++t) { zz[t] = (v8f)0.f; ss[t] = (v8f)0.f;
                                q3[t] = (v8f)0.f; qd[t] = (v8f)0.f; }
  {
    const float* xr  = x    + (row0 + n) * 64 + 2 * h;
    const float* xdr = xdot + (row0 + n) * 64 + 2 * h;
    const float* wb0 = w1   + col0 * 64 + 2 * h;
    const float* wb1 = w1   + col1 * 64 + 2 * h;
    const float* ws0 = w3s  + col0 * 64 + 2 * h;
    const float* ws1 = w3s  + col1 * 64 + 2 * h;
    #pragma unroll
    for (int k0 = 0; k0 < 64; k0 += 4) {
      v2f ax = *(const v2f*)(xr  + k0);
      v2f ad = *(const v2f*)(xdr + k0);
      v2f b0 = *(const v2f*)(wb0 + k0);
      v2f b1v= *(const v2f*)(wb1 + k0);
      v2f c0 = *(const v2f*)(ws0 + k0);
      v2f c1 = *(const v2f*)(ws1 + k0);
      zz[0] = wmma_f32_k4(ax, b0,  zz[0]);  zz[1] = wmma_f32_k4(ax, b1v, zz[1]);
      ss[0] = wmma_f32_k4(ad, b0,  ss[0]);  ss[1] = wmma_f32_k4(ad, b1v, ss[1]);
      q3[0] = wmma_f32_k4(ax, c0,  q3[0]);  q3[1] = wmma_f32_k4(ax, c1,  q3[1]);
      qd[0] = wmma_f32_k4(ad, c0,  qd[0]);  qd[1] = wmma_f32_k4(ad, c1,  qd[1]);
    }
  }
  #pragma unroll
  for (int t = 0; t < 2; ++t) {
    const int col = colbase + 16 * t + n;
    const float bi1  = b1[col];
    const float bi3s = b3s[col];
    #pragma unroll
    for (int r = 0; r < 8; ++r) {
      const int m = r + 8 * h;
      float z  = zz[t][r] + bi1;           // z = x@w1T + b1
      float yv = z * z;                    // y = z^2
      float jd = 2.0f * z * ss[t][r];      // ydot1 = 2 z s
      lds_y[m * LDS_STRIDE + col] = yv;
      lds_j[m * LDS_STRIDE + col] = jd;
      q3[t][r] += bi3s;                    // z2l3 = x@w3sT + b3s (held in regs)
    }
  }
  __syncthreads();

  // ---------------- Phase 2: layer 2, K=256, shared A for w2a & w2b ---------
  v8f z1[2], z2[2], d1[2], d2[2];
  #pragma unroll
  for (int t = 0; t < 2; ++t) { z1[t] = (v8f)0.f; z2[t] = (v8f)0.f;
                                d1[t] = (v8f)0.f; d2[t] = (v8f)0.f; }
  const float* ya = &lds_y[n * LDS_STRIDE + 2 * h];
  const float* ja = &lds_j[n * LDS_STRIDE + 2 * h];
  {
    const float* pa0 = w2a + col0 * 256 + 2 * h;
    const float* pa1 = w2a + col1 * 256 + 2 * h;
    const float* pb0 = w2b + col0 * 256 + 2 * h;
    const float* pb1 = w2b + col1 * 256 + 2 * h;
    #pragma unroll 8
    for (int k0 = 0; k0 < 256; k0 += 4) {
      v2f ay  = *(const v2f*)(ya + k0);
      v2f aj  = *(const v2f*)(ja + k0);
      v2f ba0 = *(const v2f*)(pa0 + k0);
      v2f ba1 = *(const v2f*)(pa1 + k0);
      v2f bb0 = *(const v2f*)(pb0 + k0);
      v2f bb1 = *(const v2f*)(pb1 + k0);
      z1[0] = wmma_f32_k4(ay, ba0, z1[0]);  z1[1] = wmma_f32_k4(ay, ba1, z1[1]);
      z2[0] = wmma_f32_k4(ay, bb0, z2[0]);  z2[1] = wmma_f32_k4(ay, bb1, z2[1]);
      d1[0] = wmma_f32_k4(aj, ba0, d1[0]);  d1[1] = wmma_f32_k4(aj, ba1, d1[1]);
      d2[0] = wmma_f32_k4(aj, bb0, d2[0]);  d2[1] = wmma_f32_k4(aj, bb1, d2[1]);
    }
  }
  __syncthreads();   // everyone done reading y / ydot1
  #pragma unroll
  for (int t = 0; t < 2; ++t) {
    const int col = colbase + 16 * t + n;
    const float bi2a = b2a[col];
    const float bi2b = b2b[col];
    #pragma unroll
    for (int r = 0; r < 8; ++r) {
      const int m = r + 8 * h;
      float a = z1[t][r] + bi2a;           // z1 = y@w2aT + b2a
      float b = z2[t][r] + bi2b;           // z2 = y@w2bT + b2b
      lds_y[m * LDS_STRIDE + col] = a * b;                         // y2
      lds_j[m * LDS_STRIDE + col] = a * d2[t][r] + b * d1[t][r];   // ydot2
    }
  }
  __syncthreads();

  // ---------------- Phase 3: layer 3 main path, K=256 -----------------------
  v8f z13[2], dd[2];
  #pragma unroll
  for (int t = 0; t < 2; ++t) { z13[t] = (v8f)0.f; dd[t] = (v8f)0.f; }
  {
    const float* pa0 = w3a + col0 * 256 + 2 * h;
    const float* pa1 = w3a + col1 * 256 + 2 * h;
    #pragma unroll 8
    for (int k0 = 0; k0 < 256; k0 += 4) {
      v2f ay  = *(const v2f*)(ya + k0);
      v2f aj  = *(const v2f*)(ja + k0);
      v2f ba0 = *(const v2f*)(pa0 + k0);
      v2f ba1 = *(const v2f*)(pa1 + k0);
      z13[0] = wmma_f32_k4(ay, ba0, z13[0]);  z13[1] = wmma_f32_k4(ay, ba1, z13[1]);
      dd[0]  = wmma_f32_k4(aj, ba0, dd[0]);   dd[1]  = wmma_f32_k4(aj, ba1, dd[1]);
    }
  }
  __syncthreads();   // everyone done reading y2 / ydot2
  #pragma unroll
  for (int t = 0; t < 2; ++t) {
    const int col = colbase + 16 * t + n;
    const float bi3a = b3a[col];
    #pragma unroll
    for (int r = 0; r < 8; ++r) {
      const int m = r + 8 * h;
      float a  = z13[t][r] + bi3a;                         // z1l3
      float y3 = a * q3[t][r];                             // y3 = z1l3*z2l3
      float j3 = a * qd[t][r] + q3[t][r] * dd[t][r];       // ydot3
      out_y[(row0 + m) * 256 + col] = y3;                  // output y
      lds_y[m * LDS_STRIDE + col] = y3;
      lds_j[m * LDS_STRIDE + col] = j3;
    }
  }
  __syncthreads();

  // ---------------- Phase 4: wout reductions + yy ---------------------------
  {
    const int rrow = tid >> 4;    // 0..15 (batch row in tile)
    const int part = tid & 15;    // 16-wide partial
    const int c0 = part * 16;
    float sb = 0.f, sd = 0.f;
    #pragma unroll
    for (int c = 0; c < 16; ++c) {
      float wo = wout[c0 + c];
      sb += lds_y[rrow * LDS_STRIDE + c0 + c] * wo;
      sd += lds_j[rrow * LDS_STRIDE + c0 + c] * wo;
    }
    red_b[tid] = sb;
    red_d[tid] = sd;
  }
  __syncthreads();
  if (tid < 16) {
    float s = 0.f;
    #pragma unroll
    for (int p = 0; p < 16; ++p) s += red_b[tid * 16 + p];
    out_b[row0 + tid] = s;                 // numerical_b [B,1]
  } else if (tid < 32) {
    const int r = tid - 16;
    float s = 0.f;
    #pragma unroll
    for (int p = 0; p < 16; ++p) s += red_d[r * 16 + p];
    out_bdot[row0 + r] = s;                // numerical_bdot [B]
  }
  const float s0 = scalar[0];
  for (int i = tid; i < 16 * 64; i += 256)
    out_yy[row0 * 64 + i] = s0;            // yy = broadcast scalar [B,64]
}

extern "C" void kernel_launch(void* const* d_in, const int* in_sizes, int n_in,
                              void* d_out, int out_size, void* d_ws, size_t ws_size,
                              hipStream_t stream) {
  const float* x      = (const float*)d_in[0];
  const float* xdot   = (const float*)d_in[1];
  const float* w1     = (const float*)d_in[2];
  const float* b1     = (const float*)d_in[3];
  const float* w2a    = (const float*)d_in[4];
  const float* b2a    = (const float*)d_in[5];
  const float* w2b    = (const float*)d_in[6];
  const float* b2b    = (const float*)d_in[7];
  const float* w3a    = (const float*)d_in[8];
  const float* b3a    = (const float*)d_in[9];
  const float* w3s    = (const float*)d_in[10];
  const float* b3s    = (const float*)d_in[11];
  const float* wout   = (const float*)d_in[12];
  const float* scalar = (const float*)d_in[13];

  const int Bsz = in_sizes[0] / 64;   // 4096
  float* out      = (float*)d_out;
  float* out_b    = out;                       // [B]
  float* out_bdot = out + Bsz;                 // [B]
  float* out_y    = out + 2 * Bsz;             // [B,256]
  float* out_yy   = out + 2 * Bsz + Bsz * 256; // [B,64]

  sindy_fused<<<dim3(Bsz / 16), dim3(256), 0, stream>>>(
      x, xdot, w1, b1, w2a, b2a, w2b, b2b, w3a, b3a, w3s, b3s, wout, scalar,
      out_b, out_bdot, out_y, out_yy);
}